// FastGuidedFilter_37864431682452
// MI455X (gfx1250) — compile-verified
//
#include <hip/hip_runtime.h>
#include <cstdint>

// Fast guided filter for MI455X (gfx1250), fp32.
// p_lr:(8,3,256,256) I_lr:(8,3,256,256) I_hr:(8,3,1024,1024) -> out:(8,3,1024,1024)
//
// Memory-bound (~240 MB total -> ~10us at 23.3 TB/s). No WMMA: FLOPs ~0.1G.
// CDNA5 paths used: global_load_async_to_lds_b32 + s_wait_asynccnt for the
// heavily-reused LR coefficient tiles, non-temporal B128 for the streaming
// HR tensors.

#define RADIUS 8
#define EPS 1e-4f

typedef __attribute__((ext_vector_type(4))) float v4f;

constexpr int LRW = 256;
constexpr int HRW = 1024;
constexpr int NIMG = 24;            // 8 * 3

// ------------------------- Kernel 1: LR statistics -------------------------
constexpr int TS = 32;              // output tile (LR)
constexpr int IN = TS + 2 * RADIUS; // 48, with halo

__global__ __launch_bounds__(256)
void guided_stats_kernel(const float* __restrict__ P,
                         const float* __restrict__ I,
                         float* __restrict__ Aout,
                         float* __restrict__ Bout)
{
    __shared__ float sI[IN][IN];
    __shared__ float sP[IN][IN];
    __shared__ float hI [IN][TS];
    __shared__ float hP [IN][TS];
    __shared__ float hIP[IN][TS];
    __shared__ float hII[IN][TS];

    const int tid = threadIdx.x;
    const int im  = blockIdx.z;
    const int gx0 = blockIdx.x * TS - RADIUS;
    const int gy0 = blockIdx.y * TS - RADIUS;
    const float* Ii = I + (size_t)im * LRW * LRW;
    const float* Pi = P + (size_t)im * LRW * LRW;

    // Stage input tile with zero padding (== window_sum's clipped-range sum).
    for (int i = tid; i < IN * IN; i += 256) {
        int r = i / IN, c = i % IN;
        int gy = gy0 + r, gx = gx0 + c;
        bool inb = (gx >= 0) & (gx < LRW) & (gy >= 0) & (gy < LRW);
        float vi = 0.0f, vp = 0.0f;
        if (inb) {
            vi = Ii[gy * LRW + gx];
            vp = Pi[gy * LRW + gx];
        }
        sI[r][c] = vi;
        sP[r][c] = vp;
    }
    __syncthreads();

    // Horizontal 17-tap sums of the 4 moments.
    for (int i = tid; i < IN * TS; i += 256) {
        int r = i / TS, c = i % TS;
        float si = 0.f, sp = 0.f, sip = 0.f, sii = 0.f;
        #pragma unroll
        for (int dx = 0; dx <= 2 * RADIUS; ++dx) {
            float vi = sI[r][c + dx];
            float vp = sP[r][c + dx];
            si += vi;
            sp += vp;
            sip = fmaf(vi, vp, sip);
            sii = fmaf(vi, vi, sii);
        }
        hI [r][c] = si;
        hP [r][c] = sp;
        hIP[r][c] = sip;
        hII[r][c] = sii;
    }
    __syncthreads();

    // Vertical 17-tap sums, clipped-window count, solve A,B.
    for (int i = tid; i < TS * TS; i += 256) {
        int ly = i / TS, lx = i % TS;
        float si = 0.f, sp = 0.f, sip = 0.f, sii = 0.f;
        #pragma unroll
        for (int dy = 0; dy <= 2 * RADIUS; ++dy) {
            si  += hI [ly + dy][lx];
            sp  += hP [ly + dy][lx];
            sip += hIP[ly + dy][lx];
            sii += hII[ly + dy][lx];
        }
        int y = blockIdx.y * TS + ly;
        int x = blockIdx.x * TS + lx;
        int cy = min(y + RADIUS, LRW - 1) - max(y - RADIUS, 0) + 1;
        int cx = min(x + RADIUS, LRW - 1) - max(x - RADIUS, 0) + 1;
        float inv = 1.0f / (float)(cx * cy);
        float mI  = si * inv;
        float mP  = sp * inv;
        float cov = sip * inv - mI * mP;
        float var = sii * inv - mI * mI;
        float a = cov / (var + EPS);
        float b = mP - a * mI;
        size_t o = (size_t)im * LRW * LRW + (size_t)y * LRW + x;
        Aout[o] = a;
        Bout[o] = b;
    }
}

// ---------------- Kernel 2: 4x bilinear upsample + fuse (dominant) ----------------
constexpr int OTX = 256;            // HR tile width
constexpr int OTY = 16;             // HR tile height
constexpr int LTX = OTX / 4 + 2;    // 66 LR columns (+1 halo each side)
constexpr int LTY = OTY / 4 + 2;    // 6  LR rows

__global__ __launch_bounds__(256)
void guided_upsample_kernel(const float* __restrict__ A,
                            const float* __restrict__ B,
                            const float* __restrict__ Ihr,
                            float* __restrict__ Out)
{
    __shared__ float sA[LTY][LTX];
    __shared__ float sB[LTY][LTX];

    const int tid  = threadIdx.x;
    const int im   = blockIdx.z;
    const int x0   = blockIdx.x * OTX;
    const int y0   = blockIdx.y * OTY;
    const int lrX0 = (x0 >> 2) - 1;
    const int lrY0 = (y0 >> 2) - 1;

    const float* Ai = A   + (size_t)im * LRW * LRW;
    const float* Bi = B   + (size_t)im * LRW * LRW;
    const float* Ii = Ihr + (size_t)im * HRW * HRW;
    float*       Oi = Out + (size_t)im * HRW * HRW;

    // Async-stage the A/B coefficient tile (index clamping == JAX linear
    // resize edge handling). Per-lane global->LDS DMA, no VGPR round trip.
    for (int i = tid; i < LTY * LTX; i += 256) {
        int r = i / LTX, c = i % LTX;
        int gy = min(max(lrY0 + r, 0), LRW - 1);
        int gx = min(max(lrX0 + c, 0), LRW - 1);
        const float* ga = Ai + gy * LRW + gx;
        const float* gb = Bi + gy * LRW + gx;
        uint32_t la = (uint32_t)(uintptr_t)&sA[r][c];  // flat LDS addr low 32b = LDS offset
        uint32_t lb = (uint32_t)(uintptr_t)&sB[r][c];
        asm volatile("global_load_async_to_lds_b32 %0, %1, off"
                     :: "v"(la), "v"(ga) : "memory");
        asm volatile("global_load_async_to_lds_b32 %0, %1, off"
                     :: "v"(lb), "v"(gb) : "memory");
    }
    asm volatile("s_wait_asynccnt 0" ::: "memory");
    __syncthreads();

    // 64 threads across x (float4 each = 256 px), 4 thread-rows, 4 row iters.
    const int tx = tid & 63;
    const int ty = tid >> 6;

    #pragma unroll
    for (int ry = 0; ry < OTY / 4; ++ry) {
        int y  = y0 + ry * 4 + ty;
        int ph = y & 3;                      // src = (y+0.5)/4 - 0.5 = y/4 - 0.375
        int iy = (y >> 2) - 1 + (ph >= 2 ? 1 : 0);
        float fy = (ph == 0) ? 0.625f : (ph == 1) ? 0.875f
                 : (ph == 2) ? 0.125f : 0.375f;
        int ly = iy - lrY0;                  // 0..4 within staged tile

        // y-lerp the three LR columns this thread's 4 HR pixels touch.
        float A0 = fmaf(fy, sA[ly + 1][tx    ] - sA[ly][tx    ], sA[ly][tx    ]);
        float A1 = fmaf(fy, sA[ly + 1][tx + 1] - sA[ly][tx + 1], sA[ly][tx + 1]);
        float A2 = fmaf(fy, sA[ly + 1][tx + 2] - sA[ly][tx + 2], sA[ly][tx + 2]);
        float B0 = fmaf(fy, sB[ly + 1][tx    ] - sB[ly][tx    ], sB[ly][tx    ]);
        float B1 = fmaf(fy, sB[ly + 1][tx + 1] - sB[ly][tx + 1], sB[ly][tx + 1]);
        float B2 = fmaf(fy, sB[ly + 1][tx + 2] - sB[ly][tx + 2], sB[ly][tx + 2]);

        size_t hro = (size_t)y * HRW + (size_t)(x0 + tx * 4);
        v4f iv = __builtin_nontemporal_load((const v4f*)(Ii + hro));
        v4f ov;
        // x phases 0..3: fx = .625,.875 on (c,c+1); .125,.375 on (c+1,c+2)
        {   float a = fmaf(0.625f, A1 - A0, A0);
            float b = fmaf(0.625f, B1 - B0, B0);
            ov.x = fmaf(a, iv.x, b); }
        {   float a = fmaf(0.875f, A1 - A0, A0);
            float b = fmaf(0.875f, B1 - B0, B0);
            ov.y = fmaf(a, iv.y, b); }
        {   float a = fmaf(0.125f, A2 - A1, A1);
            float b = fmaf(0.125f, B2 - B1, B1);
            ov.z = fmaf(a, iv.z, b); }
        {   float a = fmaf(0.375f, A2 - A1, A1);
            float b = fmaf(0.375f, B2 - B1, B1);
            ov.w = fmaf(a, iv.w, b); }
        __builtin_nontemporal_store(ov, (v4f*)(Oi + hro));
    }
}

extern "C" void kernel_launch(void* const* d_in, const int* in_sizes, int n_in,
                              void* d_out, int out_size, void* d_ws, size_t ws_size,
                              hipStream_t stream) {
    (void)in_sizes; (void)n_in; (void)out_size; (void)ws_size;
    const float* p_lr = (const float*)d_in[0];
    const float* I_lr = (const float*)d_in[1];
    const float* I_hr = (const float*)d_in[2];
    float* out = (float*)d_out;

    float* wsA = (float*)d_ws;                          // 24*256*256 floats
    float* wsB = wsA + (size_t)NIMG * LRW * LRW;        // 24*256*256 floats

    dim3 g1(LRW / TS, LRW / TS, NIMG);                  // 8 x 8 x 24
    guided_stats_kernel<<<g1, 256, 0, stream>>>(p_lr, I_lr, wsA, wsB);

    dim3 g2(HRW / OTX, HRW / OTY, NIMG);                // 4 x 64 x 24
    guided_upsample_kernel<<<g2, 256, 0, stream>>>(wsA, wsB, I_hr, out);
}